// actor_network_19215683682359
// MI455X (gfx1250) — compile-verified
//
#include <hip/hip_runtime.h>

// ---------------------------------------------------------------------------
// Graph net (NNConv + Graph-U-Net) for MI455X / gfx1250.
// All GEMMs use V_WMMA_F32_16X16X4_F32 (full f32 precision, matrix pipes).
// A-matrix LDS staging uses gfx1250 async LDS DMA (ASYNCcnt path).
// Working set (~52 MB scratch) fits in 192 MB L2 -> compute bound.
// Deterministic: no float atomics (per-dst edge scan in fixed order).
// ---------------------------------------------------------------------------

#define BB     128   // batch
#define NNODE  64    // nodes per graph
#define NE     512   // edges per graph
#define CF     256   // hidden width C
#define F2     32    // feature width

typedef __attribute__((ext_vector_type(2))) float v2f;
typedef __attribute__((ext_vector_type(8))) float v8f;

// ---- gfx1250 async global<->LDS DMA helpers (tracked with ASYNCcnt) -------
// Flat LDS addresses carry the LDS byte offset in the low 32 bits (ISA 10.2),
// so truncating the generic pointer yields the VDST/VSRC LDS address operand.
__device__ __forceinline__ void async_g2l_b128(void* lds_dst, const void* gsrc) {
  asm volatile("global_load_async_to_lds_b128 %0, %1, off"
               :: "v"((unsigned)(unsigned long long)lds_dst), "v"(gsrc)
               : "memory");
}
__device__ __forceinline__ void async_l2g_b128(void* gdst, const void* lds_src) {
  asm volatile("global_store_async_from_lds_b128 %0, %1, off"
               :: "v"(gdst), "v"((unsigned)(unsigned long long)lds_src)
               : "memory");
}
__device__ __forceinline__ void wait_async0() {
  asm volatile("s_wait_asynccnt 0x0" ::: "memory");
}

// ---------------------------------------------------------------------------
// Generic f32 WMMA GEMM: C(MxN) = A(MxK) @ B(KxN), 16x16 tile per wave,
// K stepped by 4 with V_WMMA_F32_16X16X4_F32. Full tiles take a check-free
// path (contiguous A-fragment loads -> b64); edge tiles use clamped-address
// unconditional loads + v_cndmask zero-fill (no exec-mask branch regions).
// Epilogue epi(row, col, acc) performs the store (and any fused math).
template <class Epi>
__device__ __forceinline__ void wmma_gemm_f32(
    const float* __restrict__ Am, int lda,
    const float* __restrict__ Bm, int ldb,
    int M, int N, int K, Epi epi)
{
  const int lane = threadIdx.x & 31;
  const int wave = threadIdx.x >> 5;
  const int h    = lane >> 4;     // half-wave (0/1)
  const int ln   = lane & 15;
  const int Mt = (M + 15) >> 4;
  const int Nt = (N + 15) >> 4;
  const int nW = blockDim.x >> 5;
  for (int t = wave; t < Mt * Nt; t += nW) {
    const int i0 = (t / Nt) << 4;
    const int j0 = (t % Nt) << 4;
    v8f acc = {};
    if (i0 + 16 <= M && j0 + 16 <= N && (K & 3) == 0) {
      // fast path: no bounds checks
      const float* ap = Am + (size_t)(i0 + ln) * lda + 2 * h;
      const float* bp = Bm + (size_t)(2 * h) * ldb + j0 + ln;
      for (int k0 = 0; k0 < K; k0 += 4) {
        v2f af, bf;
        af[0] = ap[0]; af[1] = ap[1];          // contiguous -> b64 load
        bf[0] = bp[0]; bf[1] = bp[ldb];
        ap += 4; bp += 4 * (size_t)ldb;
        acc = __builtin_amdgcn_wmma_f32_16x16x4_f32(false, af, false, bf,
                                                    (short)0, acc, false, false);
      }
    } else {
      const int mr  = i0 + ln;
      const int nc  = j0 + ln;
      const int mrc = mr < M ? mr : M - 1;     // clamped (always legal) indices
      const int ncc = nc < N ? nc : N - 1;
      for (int k0 = 0; k0 < K; k0 += 4) {
        v2f af, bf;
#pragma unroll
        for (int v = 0; v < 2; ++v) {
          const int ka  = k0 + v + 2 * h;      // ISA 7.12.2 f32 A/B layout
          const int kac = ka < K ? ka : K - 1;
          const float av = Am[(size_t)mrc * lda + kac];   // unconditional load
          const float bv = Bm[(size_t)kac * ldb + ncc];   // unconditional load
          af[v] = (mr < M && ka < K) ? av : 0.0f;         // v_cndmask select
          bf[v] = (nc < N && ka < K) ? bv : 0.0f;
        }
        acc = __builtin_amdgcn_wmma_f32_16x16x4_f32(false, af, false, bf,
                                                    (short)0, acc, false, false);
      }
    }
#pragma unroll
    for (int r = 0; r < 8; ++r) {
      const int row = i0 + r + 8 * h;
      const int col = j0 + ln;
      if (row < M && col < N) epi(row, col, acc[r]);
    }
  }
}

// ---------------------------------------------------------------------------
// Adjacency: A[b][dst][src] = 1 (self loops removed).
__global__ __launch_bounds__(256) void k_adj(const int* __restrict__ ei,
                                             float* __restrict__ A)
{
  const int b = blockIdx.x;
  float* Ab = A + b * NNODE * NNODE;
  for (int i = threadIdx.x; i < NNODE * NNODE; i += 256) Ab[i] = 0.0f;
  __syncthreads();
  const int* srcs = ei + b * 2 * NE;
  const int* dsts = srcs + NE;
  for (int e = threadIdx.x; e < NE; e += 256) {
    const int s = srcs[e], d = dsts[e];
    if (s != d) Ab[d * NNODE + s] = 1.0f;   // duplicates write same value
  }
}

// ---------------------------------------------------------------------------
// NNConv edge aggregation, deterministic: one wave per destination node,
// scans edges in index order; lane = output channel (F2 = 32 = wave32).
// msg[e,o] = sum_i x[src[e],i] * relu(ea[e]*mw[i,o] + mb[i,o])
__global__ __launch_bounds__(256) void k_edge_agg(
    const float* __restrict__ x, int ldx, int fin,
    const int* __restrict__ ei, const float* __restrict__ ea,
    const float* __restrict__ mw, const float* __restrict__ mb,
    float* __restrict__ sums, float* __restrict__ cnt)
{
  const int b    = blockIdx.x;
  const int lane = threadIdx.x & 31;
  const int wave = threadIdx.x >> 5;
  const int* srcs = ei + b * 2 * NE;
  const int* dsts = srcs + NE;
  const float* eab = ea + b * NE;
  for (int d = wave; d < NNODE; d += 8) {
    float acc = 0.0f, c = 0.0f;
    for (int e = 0; e < NE; ++e) {
      if (dsts[e] == d) {                       // uniform across the wave
        c += 1.0f;
        const float ev = eab[e];
        const float* xs = x + (b * NNODE + srcs[e]) * ldx;
        for (int i = 0; i < fin; ++i) {
          const float wv = fmaxf(ev * mw[i * F2 + lane] + mb[i * F2 + lane], 0.0f);
          acc += xs[i] * wv;
        }
      }
    }
    sums[(b * NNODE + d) * F2 + lane] = acc;
    if (lane == 0) cnt[b * NNODE + d] = c;
  }
}

// out = relu(sums/max(cnt,1) + x@rw + bias)   (WMMA: 64 x 32 x fin per graph)
__global__ __launch_bounds__(256) void k_xrw_combine(
    const float* __restrict__ x, int ldx, int fin,
    const float* __restrict__ rw, const float* __restrict__ bias,
    const float* __restrict__ sums, const float* __restrict__ cnt,
    float* __restrict__ out)
{
  const int b = blockIdx.x;
  const float* xb = x + b * NNODE * ldx;
  const float* sb = sums + b * NNODE * F2;
  const float* cb = cnt + b * NNODE;
  float* ob = out + b * NNODE * F2;
  __builtin_prefetch(rw + threadIdx.x, 0, 1);
  wmma_gemm_f32(xb, ldx, rw, F2, NNODE, F2, fin,
    [&](int r, int c, float v) {
      // v_rcp_f32 instead of full IEEE divide: cnt is a small integer count
      const float o = sb[r * F2 + c] * __builtin_amdgcn_rcpf(fmaxf(cb[r], 1.0f))
                      + v + bias[c];
      ob[r * F2 + c] = fmaxf(o, 0.0f);
    });
}

// ---------------------------------------------------------------------------
// Full Graph-U-Net for one graph per workgroup (256 thr = 8 waves).
// A (64x64, zero padded) lives in LDS; features/skips in global scratch.
__global__ __launch_bounds__(256) void k_unet(
    const float* __restrict__ xin,   // B*64*32
    const float* __restrict__ Abr,   // B*64*64 (original A, read only)
    const float* __restrict__ dw0, const float* __restrict__ db0,
    const float* __restrict__ dws, const float* __restrict__ dbs,
    const float* __restrict__ pws,
    const float* __restrict__ uws, const float* __restrict__ ubs,
    const float* __restrict__ uwl, const float* __restrict__ ubl,
    float* __restrict__ xcur_g, float* __restrict__ xw_g,
    float* __restrict__ xsv_g, float* __restrict__ Asv_g,
    float* __restrict__ xout)        // B*64*32
{
  __shared__ __align__(16) float sA[NNODE * NNODE];
  __shared__ __align__(16) float sT[NNODE * NNODE];
  __shared__ float sd[NNODE];
  __shared__ float ss[NNODE];
  __shared__ float ssv[NNODE];
  __shared__ int   sp[NNODE];
  __shared__ int   spsv[3][NNODE];
  __shared__ float sred[256];

  const int b = blockIdx.x;
  const int tid = threadIdx.x;
  float* xc  = xcur_g + b * NNODE * CF;
  float* xw  = xw_g   + b * NNODE * CF;
  float* xs0 = xsv_g + (0 * BB + b) * NNODE * CF;
  float* xs1 = xsv_g + (1 * BB + b) * NNODE * CF;
  float* xs2 = xsv_g + (2 * BB + b) * NNODE * CF;
  float* As0 = Asv_g + (0 * BB + b) * NNODE * NNODE;  // A after pool level0 (52)
  float* As1 = Asv_g + (1 * BB + b) * NNODE * NNODE;  // A after pool level1 (42)

  // async LDS DMA staging of the 64x64 A matrix (16 KB, 4 x b128 per thread)
  auto loadA = [&](const float* gsrc) {
    for (int i = tid * 4; i < NNODE * NNODE; i += 1024)
      async_g2l_b128(&sA[i], gsrc + i);
    wait_async0();
    __syncthreads();
  };
  auto storeA = [&](float* gdst) {
    for (int i = tid * 4; i < NNODE * NNODE; i += 1024)
      async_l2g_b128(gdst + i, &sA[i]);
    wait_async0();
    __syncthreads();
  };
  auto copyx = [&](float* dst, const float* srcp, int rows) {
    for (int i = tid; i < rows * CF; i += 256) dst[i] = srcp[i];
    __syncthreads();
  };

  // GCN: out = An @ (x@W) + b ; An = dis_i*(A+2I)*dis_j, dis = rsqrt(rowsum+2)
  auto gcn = [&](const float* xsrc, int ldx, int n, int Cin, int Cout,
                 const float* W, const float* bias, bool dorelu,
                 float* xdst, int ldd) {
    __builtin_prefetch(W + tid, 0, 1);
    if (tid < NNODE) {
      float dsum = 2.0f;
      for (int j = 0; j < NNODE; ++j) dsum += sA[tid * NNODE + j];
      sd[tid] = dsum > 0.0f ? rsqrtf(dsum) : 0.0f;
    }
    __syncthreads();
    // xw2[j] = dis[j] * (x@W)[j]
    wmma_gemm_f32(xsrc, ldx, W, Cout, n, Cout, Cin,
      [&](int r, int c, float v) { xw[r * CF + c] = v * sd[r]; });
    __syncthreads();
    // out[i] = dis[i]*( (A@xw2)[i] + 2*xw2[i] ) + b
    wmma_gemm_f32(sA, NNODE, xw, CF, n, Cout, n,
      [&](int r, int c, float v) {
        const float o = sd[r] * (v + 2.0f * xw[r * CF + c]) + bias[c];
        xdst[r * ldd + c] = dorelu ? fmaxf(o, 0.0f) : o;
      });
    __syncthreads();
  };

  // A <- (A+I)@(A+I) with zero diagonal = A@A + 2A (+I), diag -> 0
  auto augment = [&]() {
    wmma_gemm_f32(sA, NNODE, sA, NNODE, NNODE, NNODE, NNODE,
      [&](int r, int c, float v) {
        sT[r * NNODE + c] = (r == c) ? 0.0f : (v + 2.0f * sA[r * NNODE + c]);
      });
    __syncthreads();
    for (int i = tid; i < NNODE * NNODE; i += 256) sA[i] = sT[i];
    __syncthreads();
  };

  // top-k pooling: stable rank (matches jax.lax.top_k tie ordering)
  auto pool = [&](int n, int k, const float* pw) {
    sred[tid] = pw[tid] * pw[tid];
    __syncthreads();
    for (int s = 128; s > 0; s >>= 1) {
      if (tid < s) sred[tid] += sred[tid + s];
      __syncthreads();
    }
    const float wn = sqrtf(sred[0]);
    __syncthreads();
    if (tid < n) {
      float a = 0.0f;
      const float* xr = xc + tid * CF;
      for (int c = 0; c < CF; ++c) a += xr[c] * pw[c];
      ss[tid] = tanhf(a / wn);
    }
    __syncthreads();
    if (tid < n) {
      const float si = ss[tid];
      int rk = 0;
      for (int j = 0; j < n; ++j) {
        const float sj = ss[j];
        if (sj > si || (sj == si && j < tid)) ++rk;
      }
      if (rk < k) { sp[rk] = tid; ssv[rk] = si; }
    }
    __syncthreads();
    // x <- x[perm] * sv   (via xw temp; perm entries unique)
    for (int i = tid; i < k * CF; i += 256) {
      const int r = i >> 8, c = i & (CF - 1);
      xw[i] = xc[sp[r] * CF + c] * ssv[r];
    }
    __syncthreads();
    for (int i = tid; i < k * CF; i += 256) xc[i] = xw[i];
    // A <- A[perm][:,perm], zero padded to 64x64
    for (int i = tid; i < NNODE * NNODE; i += 256) {
      const int r = i >> 6, q = i & 63;
      sT[i] = (r < k && q < k) ? sA[sp[r] * NNODE + sp[q]] : 0.0f;
    }
    __syncthreads();
    for (int i = tid; i < NNODE * NNODE; i += 256) sA[i] = sT[i];
    __syncthreads();
  };

  // x <- res + scatter(x, perm)
  auto unpool = [&](const float* res, int n_to, const int* pv, int n_from) {
    for (int i = tid; i < n_to * CF; i += 256) xw[i] = res[i];
    __syncthreads();
    for (int i = tid; i < n_from * CF; i += 256) {
      const int r = i >> 8, c = i & (CF - 1);
      xw[pv[r] * CF + c] += xc[i];
    }
    __syncthreads();
    for (int i = tid; i < n_to * CF; i += 256) xc[i] = xw[i];
    __syncthreads();
  };

  // ---------------- down path ----------------
  loadA(Abr + b * NNODE * NNODE);
  gcn(xin + b * NNODE * F2, F2, NNODE, F2, CF, dw0, db0, true, xc, CF);
  copyx(xs0, xc, NNODE);

  // level 0: 64 -> 52
  augment();
  pool(64, 52, pws + 0 * CF);
  if (tid < 52) spsv[0][tid] = sp[tid];
  storeA(As0);
  gcn(xc, CF, 52, CF, CF, dws + 0 * CF * CF, dbs + 0 * CF, true, xc, CF);
  copyx(xs1, xc, 52);

  // level 1: 52 -> 42
  augment();
  pool(52, 42, pws + 1 * CF);
  if (tid < 42) spsv[1][tid] = sp[tid];
  storeA(As1);
  gcn(xc, CF, 42, CF, CF, dws + 1 * CF * CF, dbs + 1 * CF, true, xc, CF);
  copyx(xs2, xc, 42);

  // level 2: 42 -> 34
  augment();
  pool(42, 34, pws + 2 * CF);
  if (tid < 34) spsv[2][tid] = sp[tid];
  gcn(xc, CF, 34, CF, CF, dws + 2 * CF * CF, dbs + 2 * CF, true, xc, CF);

  // ---------------- up path ----------------
  // up 0: 34 -> 42 (skip xs2, A after pool1)
  unpool(xs2, 42, spsv[2], 34);
  loadA(As1);
  gcn(xc, CF, 42, CF, CF, uws + 0 * CF * CF, ubs + 0 * CF, true, xc, CF);
  // up 1: 42 -> 52 (skip xs1, A after pool0)
  unpool(xs1, 52, spsv[1], 42);
  loadA(As0);
  gcn(xc, CF, 52, CF, CF, uws + 1 * CF * CF, ubs + 1 * CF, true, xc, CF);
  // up 2: 52 -> 64 (skip xs0, original A, 256->32; branch-level relu fused)
  unpool(xs0, 64, spsv[0], 52);
  loadA(Abr + b * NNODE * NNODE);
  gcn(xc, CF, 64, CF, F2, uwl, ubl, true, xout + b * NNODE * F2, F2);
}

// ---------------------------------------------------------------------------
// mean over nodes + pack [r(32) | jw(16)] into the 96-wide head input.
__global__ __launch_bounds__(64) void k_mean_pack(
    const float* __restrict__ x, const float* __restrict__ jw,
    float* __restrict__ cvec, int base)
{
  const int b = blockIdx.x, t = threadIdx.x;
  if (t < F2) {
    float s = 0.0f;
    for (int i = 0; i < NNODE; ++i) s += x[(b * NNODE + i) * F2 + t];
    cvec[b * 96 + base + t] = s * (1.0f / (float)NNODE);
  } else if (t < 48) {
    cvec[b * 96 + base + F2 + (t - F2)] = jw[b * 16 + (t - F2)];
  }
}

// head: out(128x256) = relu(c(128x96) @ w(96x256) + b)
__global__ __launch_bounds__(256) void k_head(
    const float* __restrict__ cvec, const float* __restrict__ w,
    const float* __restrict__ bias, float* __restrict__ out)
{
  const int m0 = blockIdx.x * 16;
  wmma_gemm_f32(cvec + m0 * 96, 96, w, 256, 16, 256, 96,
    [&](int r, int c, float v) {
      out[(m0 + r) * 256 + c] = fmaxf(v + bias[c], 0.0f);
    });
}

// ---------------------------------------------------------------------------
extern "C" void kernel_launch(void* const* d_in, const int* in_sizes, int n_in,
                              void* d_out, int out_size, void* d_ws, size_t ws_size,
                              hipStream_t stream)
{
  (void)in_sizes; (void)n_in; (void)out_size; (void)ws_size;

  const float* x1  = (const float*)d_in[0];
  const int*   ei1 = (const int*)  d_in[1];
  const float* ea1 = (const float*)d_in[2];
  const float* jw1 = (const float*)d_in[3];
  const float* x2  = (const float*)d_in[4];
  const int*   ei2 = (const int*)  d_in[5];
  const float* ea2 = (const float*)d_in[6];
  const float* jw2 = (const float*)d_in[7];
  const float* m1w = (const float*)d_in[8];
  const float* m1b = (const float*)d_in[9];
  const float* r1w = (const float*)d_in[10];
  const float* c1b = (const float*)d_in[11];
  const float* m2w = (const float*)d_in[12];
  const float* m2b = (const float*)d_in[13];
  const float* r2w = (const float*)d_in[14];
  const float* c2b = (const float*)d_in[15];
  // u1: d_in[16..24], u2: d_in[25..33]  (dw0,db0,dws,dbs,pws,uws,ubs,uwl,ubl)
  const float* bbw = (const float*)d_in[34];
  const float* bbb = (const float*)d_in[35];

  // workspace layout (floats)
  float* w = (float*)d_ws;
  size_t o = 0;
  float* Abuf = w + o; o += (size_t)BB * NNODE * NNODE;     // 524288
  float* sums = w + o; o += (size_t)BB * NNODE * F2;        // 262144 (cnt follows)
  float* cnt  = w + o; o += (size_t)BB * NNODE;             // 8192
  float* xio  = w + o; o += (size_t)BB * NNODE * F2;
  float* xio2 = w + o; o += (size_t)BB * NNODE * F2;
  float* xc   = w + o; o += (size_t)BB * NNODE * CF;
  float* xww  = w + o; o += (size_t)BB * NNODE * CF;
  float* xsv  = w + o; o += (size_t)3 * BB * NNODE * CF;
  float* Asv  = w + o; o += (size_t)2 * BB * NNODE * NNODE;
  float* cvec = w + o; o += (size_t)BB * 96;                // total ~12.9M floats

  auto run_branch = [&](const float* x0, const int* ei, const float* ea,
                        const float* jwp, int base) {
    k_adj<<<BB, 256, 0, stream>>>(ei, Abuf);
    // NNConv 1 (fin=64)
    k_edge_agg<<<BB, 256, 0, stream>>>(x0, 64, 64, ei, ea, m1w, m1b, sums, cnt);
    k_xrw_combine<<<BB, 256, 0, stream>>>(x0, 64, 64, r1w, c1b, sums, cnt, xio);
    // U-Net 1
    k_unet<<<BB, 256, 0, stream>>>(xio, Abuf,
        (const float*)d_in[16], (const float*)d_in[17], (const float*)d_in[18],
        (const float*)d_in[19], (const float*)d_in[20], (const float*)d_in[21],
        (const float*)d_in[22], (const float*)d_in[23], (const float*)d_in[24],
        xc, xww, xsv, Asv, xio2);
    // NNConv 2 (fin=32)
    k_edge_agg<<<BB, 256, 0, stream>>>(xio2, 32, 32, ei, ea, m2w, m2b, sums, cnt);
    k_xrw_combine<<<BB, 256, 0, stream>>>(xio2, 32, 32, r2w, c2b, sums, cnt, xio);
    // U-Net 2
    k_unet<<<BB, 256, 0, stream>>>(xio, Abuf,
        (const float*)d_in[25], (const float*)d_in[26], (const float*)d_in[27],
        (const float*)d_in[28], (const float*)d_in[29], (const float*)d_in[30],
        (const float*)d_in[31], (const float*)d_in[32], (const float*)d_in[33],
        xc, xww, xsv, Asv, xio2);
    k_mean_pack<<<BB, 64, 0, stream>>>(xio2, jwp, cvec, base);
  };

  run_branch(x1, ei1, ea1, jw1, 0);
  run_branch(x2, ei2, ea2, jw2, 48);
  k_head<<<8, 256, 0, stream>>>(cvec, bbw, bbb, (float*)d_out);
}